// CustomLossCNN1DFast_67894843015536
// MI455X (gfx1250) — compile-verified
//
#include <hip/hip_runtime.h>

// ---------------------------------------------------------------------------
// Problem constants (fixed by the reference)
// ---------------------------------------------------------------------------
#define NDIM (64 * 64 * 64)   // 262144 unknowns
#define BSZ 128               // batch of residual vectors
#define NNZT (7 * NDIM)       // 1,835,008 nonzeros
#define CAP 32                // per-row bucket capacity (Poisson(7): P(>32) ~ 0)

typedef __attribute__((ext_vector_type(2))) float v2f;
typedef __attribute__((ext_vector_type(8))) float v8f;

// ---------------------------------------------------------------------------
// K0: zero the row counters and the 5x128 dot accumulators
// ---------------------------------------------------------------------------
__global__ void k_zero(int* __restrict__ cnt, float* __restrict__ dots) {
    int i = blockIdx.x * 256 + threadIdx.x;
    if (i < NDIM) cnt[i] = 0;
    if (i < 5 * BSZ) dots[i] = 0.0f;
}

// ---------------------------------------------------------------------------
// K1: transpose (BS,N) row-major -> (N,BS) so that the 128-wide batch is the
// fast (contiguous, 512B) dimension for the SpMM gather/scatter.
// LDS tile 32(n) x 128(b), padded row stride 129 to kill bank conflicts.
// ---------------------------------------------------------------------------
__global__ __launch_bounds__(256) void k_transpose(
    const float* __restrict__ yp, const float* __restrict__ yt,
    float* __restrict__ yT, float* __restrict__ tT) {
    __shared__ float tile[32][129];
    const int n0 = blockIdx.x * 32;
    const int t  = threadIdx.x;
    const int nl = t & 31;    // load phase: contiguous along n
    const int bq = t >> 5;    // 8 b-values per pass

    // ---- y_pred ----
#pragma unroll
    for (int p = 0; p < 16; ++p) {
        int b = bq + p * 8;
        tile[nl][b] = yp[(size_t)b * NDIM + n0 + nl];
    }
    __syncthreads();
#pragma unroll
    for (int p = 0; p < 16; ++p) {
        int n2 = (t >> 7) + p * 2;   // 2 n-rows per pass
        int b  = t & 127;            // contiguous along b
        yT[(size_t)(n0 + n2) * BSZ + b] = tile[n2][b];
    }
    __syncthreads();

    // ---- y_true ----
#pragma unroll
    for (int p = 0; p < 16; ++p) {
        int b = bq + p * 8;
        tile[nl][b] = yt[(size_t)b * NDIM + n0 + nl];
    }
    __syncthreads();
#pragma unroll
    for (int p = 0; p < 16; ++p) {
        int n2 = (t >> 7) + p * 2;
        int b  = t & 127;
        tT[(size_t)(n0 + n2) * BSZ + b] = tile[n2][b];
    }
}

// ---------------------------------------------------------------------------
// K2: bucket the COO entries by output row (int atomics only; the expensive
// 235M float scatter-atomics of the naive formulation are eliminated).
// ---------------------------------------------------------------------------
__global__ void k_bucket(const int* __restrict__ rows,
                         int* __restrict__ cnt, int* __restrict__ buck) {
    int j = blockIdx.x * 256 + threadIdx.x;
    if (j < NNZT) {
        int r = rows[j];
        int p = atomicAdd(&cnt[r], 1);
        if (p < CAP) buck[r * CAP + p] = j;
    }
}

// ---------------------------------------------------------------------------
// K3: wave-per-row SpMM. Each wave32 owns one output row r: it rank-orders its
// bucket entries by original nnz index (deterministic summation order), then
// for each nnz gathers the 512B column yT[c,:] (float4/lane) and FMAs into a
// register accumulator. One non-atomic 512B store per row.
// ---------------------------------------------------------------------------
__global__ __launch_bounds__(256) void k_spmm(
    const float* __restrict__ vals, const int* __restrict__ cols,
    const int* __restrict__ cnt, const int* __restrict__ buck,
    const float* __restrict__ yT, float* __restrict__ hT) {
    const int lane = threadIdx.x & 31;
    const int wave = threadIdx.x >> 5;
    const int r    = blockIdx.x * 8 + wave;

    int k = cnt[r];
    if (k > CAP) k = CAP;

    int myIdx = (lane < k) ? buck[r * CAP + lane] : 0x7fffffff;

    // rank of this lane's entry = #entries with smaller original index
    int rank = 0;
    for (int j = 0; j < k; ++j) {
        int other = __shfl(myIdx, j, 32);
        rank += (other < myIdx) ? 1 : 0;
    }

    float4 acc = make_float4(0.f, 0.f, 0.f, 0.f);
    for (int s = 0; s < k; ++s) {
        unsigned long long m = __ballot(lane < k && rank == s);
        int src = (int)__builtin_ctzll(m);
        int jj  = __shfl(myIdx, src, 32);
        float v = vals[jj];
        int   c = cols[jj];
        const float4* col = (const float4*)(yT + (size_t)c * BSZ);
        float4 x = col[lane];
        acc.x = fmaf(v, x.x, acc.x);
        acc.y = fmaf(v, x.y, acc.y);
        acc.z = fmaf(v, x.z, acc.z);
        acc.w = fmaf(v, x.w, acc.w);
    }
    ((float4*)(hT + (size_t)r * BSZ))[lane] = acc;
}

// ---------------------------------------------------------------------------
// K4: the five per-batch reductions as diagonal blocks of X^T Y, contracted
// over N with V_WMMA_F32_16X16X4_F32 (full fp32 — matches reference
// precision). For the f32 16x16x4 layout the A-fragment of X^T and the
// B-fragment of X are bitwise identical, so one 3-fragment load (t,p,h)
// feeds 5 WMMAs: TP, PH, TT, TH, HH.
//   fragment: lanes 0-15 hold (n0+0,n0+1), lanes 16-31 hold (n0+2,n0+3),
//             column b0 + (lane&15).
// Diagonal of a 16x16 f32 C tile lives in lanes 0-7 (vgpr=lane) and
// lanes 24-31 (vgpr=lane-24).
// ---------------------------------------------------------------------------
#define DOT_BLOCKS 512
__global__ __launch_bounds__(256) void k_dots(
    const float* __restrict__ yT, const float* __restrict__ tT,
    const float* __restrict__ hT, float* __restrict__ dots) {
    const int lane = threadIdx.x & 31;
    const int wave = threadIdx.x >> 5;   // 0..7 -> diagonal 16x16 tile
    const int b0   = wave * 16;
    const int half = lane >> 4;          // 0: K=0,1   1: K=2,3
    const int l16  = lane & 15;

    const int chunk  = NDIM / DOT_BLOCKS;          // 512 n per block
    const int nStart = blockIdx.x * chunk;

    v8f cTP = {}, cPH = {}, cTT = {}, cTH = {}, cHH = {};

    for (int n0 = nStart; n0 < nStart + chunk; n0 += 4) {
        size_t base = (size_t)(n0 + 2 * half) * BSZ + b0 + l16;
        v2f fT, fP, fH;
        fT.x = tT[base]; fT.y = tT[base + BSZ];
        fP.x = yT[base]; fP.y = yT[base + BSZ];
        fH.x = hT[base]; fH.y = hT[base + BSZ];

        cTP = __builtin_amdgcn_wmma_f32_16x16x4_f32(false, fT, false, fP, (short)0, cTP, false, false);
        cPH = __builtin_amdgcn_wmma_f32_16x16x4_f32(false, fP, false, fH, (short)0, cPH, false, false);
        cTT = __builtin_amdgcn_wmma_f32_16x16x4_f32(false, fT, false, fT, (short)0, cTT, false, false);
        cTH = __builtin_amdgcn_wmma_f32_16x16x4_f32(false, fT, false, fH, (short)0, cTH, false, false);
        cHH = __builtin_amdgcn_wmma_f32_16x16x4_f32(false, fH, false, fH, (short)0, cHH, false, false);
    }

    // extract the diagonal element this lane owns (if any) and accumulate
    const int col   = lane & 15;
    const bool valid = (lane < 8) || (lane >= 24);
    float dTP = 0.f, dPH = 0.f, dTT = 0.f, dTH = 0.f, dHH = 0.f;
#pragma unroll
    for (int v = 0; v < 8; ++v) {
        int m = v + ((lane >= 16) ? 8 : 0);   // row of C held in vgpr v
        if (m == col) { dTP = cTP[v]; dPH = cPH[v]; dTT = cTT[v]; dTH = cTH[v]; dHH = cHH[v]; }
    }
    if (valid) {
        int b = b0 + col;
        atomicAdd(&dots[0 * BSZ + b], dTP);
        atomicAdd(&dots[1 * BSZ + b], dPH);
        atomicAdd(&dots[2 * BSZ + b], dTT);
        atomicAdd(&dots[3 * BSZ + b], dTH);
        atomicAdd(&dots[4 * BSZ + b], dHH);
    }
}

// ---------------------------------------------------------------------------
// K5: loss = mean_b [ TT - 2 s TH + s^2 HH ],  s = TP / PH
// (closed-form expansion of ||y_true - s*Yhat||^2; no residual pass needed)
// ---------------------------------------------------------------------------
__global__ void k_final(const float* __restrict__ dots, float* __restrict__ out) {
    __shared__ float red[BSZ];
    int b = threadIdx.x;
    float tp = dots[0 * BSZ + b];
    float ph = dots[1 * BSZ + b];
    float tt = dots[2 * BSZ + b];
    float th = dots[3 * BSZ + b];
    float hh = dots[4 * BSZ + b];
    float s = tp / ph;
    red[b] = tt - 2.0f * s * th + s * s * hh;
    __syncthreads();
    for (int off = BSZ / 2; off > 0; off >>= 1) {
        if (b < off) red[b] += red[b + off];
        __syncthreads();
    }
    if (b == 0) out[0] = red[0] / (float)BSZ;
}

// ---------------------------------------------------------------------------
// launch
// ---------------------------------------------------------------------------
extern "C" void kernel_launch(void* const* d_in, const int* in_sizes, int n_in,
                              void* d_out, int out_size, void* d_ws, size_t ws_size,
                              hipStream_t stream) {
    const float* y_pred = (const float*)d_in[0];   // (BS, N) f32
    const float* y_true = (const float*)d_in[1];   // (BS, N) f32
    const float* vals   = (const float*)d_in[2];   // (NNZ,)  f32
    const int*   rows   = (const int*)d_in[3];     // (NNZ,)  i32
    const int*   cols   = (const int*)d_in[4];     // (NNZ,)  i32
    float* out = (float*)d_out;

    // workspace carve-up (~437 MB)
    char* ws = (char*)d_ws;
    size_t off = 0;
    float* yT   = (float*)(ws + off); off += (size_t)NDIM * BSZ * sizeof(float);
    float* tT   = (float*)(ws + off); off += (size_t)NDIM * BSZ * sizeof(float);
    float* hT   = (float*)(ws + off); off += (size_t)NDIM * BSZ * sizeof(float);
    int*   cnt  = (int*)  (ws + off); off += (size_t)NDIM * sizeof(int);
    int*   buck = (int*)  (ws + off); off += (size_t)NDIM * CAP * sizeof(int);
    float* dots = (float*)(ws + off); off += 5 * BSZ * sizeof(float);

    (void)in_sizes; (void)n_in; (void)out_size; (void)ws_size;

    k_zero<<<(NDIM + 255) / 256, 256, 0, stream>>>(cnt, dots);
    k_transpose<<<NDIM / 32, 256, 0, stream>>>(y_pred, y_true, yT, tT);
    k_bucket<<<(NNZT + 255) / 256, 256, 0, stream>>>(rows, cnt, buck);
    k_spmm<<<NDIM / 8, 256, 0, stream>>>(vals, cols, cnt, buck, yT, hT);
    k_dots<<<DOT_BLOCKS, 256, 0, stream>>>(yT, tT, hT, dots);
    k_final<<<1, BSZ, 0, stream>>>(dots, out);
}